// SSD_15350213116147
// MI455X (gfx1250) — compile-verified
//
#include <hip/hip_runtime.h>
#include <cstdint>

// SSD post-process decode for MI455X (gfx1250, wave32).
// Memory-bound (AI ~0.36 FLOP/B): optimize the data path, not the math.
//  - Confidences staged to LDS via CDNA5 async global->LDS B128 copies
//    (coalesced 512B/instr per wave, ASYNCcnt-tracked, no VGPR round trip).
//    Copy loop is fully unrolled: 5 full rounds + 64-lane tail, so all 6
//    async ops issue back-to-back with no exec-mask loop machinery.
//  - Offsets / default boxes loaded as float4 BEFORE s_wait_asynccnt, so
//    the async pipeline overlaps the regular LOADcnt traffic.
//  - LDS readback at dword stride 21 (coprime with 64 banks -> conflict-free).
//  - Default RT cache hints: the 139MB working set fits the 192MB L2, so
//    graph replays can run L2-resident (no NT/LU streaming hints).

#define BATCH    128
#define NBOX     8732
#define NCLS     21
#define TILE     256
#define CONF_TH  0.05f

#if defined(__HIP_DEVICE_COMPILE__) && defined(__gfx1250__) && \
    __has_builtin(__builtin_amdgcn_global_load_async_to_lds_b128)
#define USE_ASYNC_LDS 1
#else
#define USE_ASYNC_LDS 0
#endif

#if USE_ASYNC_LDS
typedef int v4i __attribute__((ext_vector_type(4)));
typedef __attribute__((address_space(1))) v4i* g_v4i_ptr;   // global (AS1)
typedef __attribute__((address_space(3))) v4i* l_v4i_ptr;   // LDS (AS3)
#endif

__global__ __launch_bounds__(TILE) void ssd_decode_kernel(
    const float* __restrict__ off,    // [B, N, 4]
    const float* __restrict__ conf,   // [B, N, 21]
    const float* __restrict__ dbox,   // [N, 4]  (cx, cy, w, h)
    float* __restrict__ out_boxes,    // [B, N, 4]
    float* __restrict__ out_cls,      // [B, N]  (stored as float)
    float* __restrict__ out_valid)    // [B, N]  (stored as float 0/1)
{
    __shared__ __align__(16) float sconf[TILE * NCLS];   // 21504 B

    const int          tid = threadIdx.x;
    const unsigned int g0  = (unsigned int)blockIdx.x * TILE;   // first anchor of block
    const unsigned int g   = g0 + tid;                          // this thread's flat anchor

    // ---- Stage the block's contiguous confidence chunk into LDS ----------
    // 256 anchors * 21 classes * 4B = 21504 B = 1344 x b128 = 5*256 + 64.
    const float* csrc = conf + (size_t)g0 * NCLS;               // 16B aligned (g0*84B)
#if USE_ASYNC_LDS
    {
#pragma unroll
        for (int r = 0; r < 5; ++r) {
            const int i = tid + r * TILE;
            __builtin_amdgcn_global_load_async_to_lds_b128(
                (g_v4i_ptr)(uintptr_t)(csrc + (size_t)i * 4),
                (l_v4i_ptr)(&sconf[i * 4]),
                /*offset=*/0, /*cpol=*/0);
        }
        if (tid < 64) {                                          // 64-lane tail round
            const int i = tid + 5 * TILE;
            __builtin_amdgcn_global_load_async_to_lds_b128(
                (g_v4i_ptr)(uintptr_t)(csrc + (size_t)i * 4),
                (l_v4i_ptr)(&sconf[i * 4]),
                /*offset=*/0, /*cpol=*/0);
        }
    }
#else
    for (int i = tid; i < TILE * NCLS; i += TILE) sconf[i] = csrc[i];
#endif

    // ---- Issue the per-thread float4 loads while the async copies fly ----
    const unsigned int n = g % NBOX;            // anchor index within batch
    const float4 d = ((const float4*)dbox)[n];
    const float4 o = ((const float4*)off)[g];

#if USE_ASYNC_LDS
#if __has_builtin(__builtin_amdgcn_s_wait_asynccnt)
    __builtin_amdgcn_s_wait_asynccnt(0);
#else
    asm volatile("s_wait_asynccnt 0" ::: "memory");
#endif
#endif
    __syncthreads();

    // ---- Per-anchor argmax / max over 21 classes (LDS, conflict-free) ----
    const float* my = &sconf[tid * NCLS];
    float maxv = my[0];
    int   maxc = 0;
#pragma unroll
    for (int c = 1; c < NCLS; ++c) {
        float v = my[c];
        if (v > maxv) { maxv = v; maxc = c; }   // strict '>' keeps first index on ties
    }
    const bool valid = (maxc != 0) && (maxv > CONF_TH);

    // ---- Box decode (float4 coalesced) -----------------------------------
    const float cx = fmaf(o.x, d.z, d.x);
    const float cy = fmaf(o.y, d.w, d.y);
    const float w  = d.z * __expf(o.z);
    const float h  = d.w * __expf(o.w);

    float4 box;
    box.x = cx - 0.5f * w;
    box.y = cy - 0.5f * h;
    box.z = cx + 0.5f * w;
    box.w = cy + 0.5f * h;
    if (!valid) { box.x = 0.f; box.y = 0.f; box.z = 0.f; box.w = 0.f; }

    ((float4*)out_boxes)[g] = box;
    out_cls[g]   = (float)maxc;
    out_valid[g] = valid ? 1.0f : 0.0f;
}

extern "C" void kernel_launch(void* const* d_in, const int* in_sizes, int n_in,
                              void* d_out, int out_size, void* d_ws, size_t ws_size,
                              hipStream_t stream) {
    (void)in_sizes; (void)n_in; (void)out_size; (void)d_ws; (void)ws_size;

    const float* off  = (const float*)d_in[0];   // [B, N, 4]
    const float* conf = (const float*)d_in[1];   // [B, N, 21]
    const float* dbox = (const float*)d_in[2];   // [N, 4]

    const size_t NB = (size_t)BATCH * NBOX;      // 1,117,696 anchors
    float* out       = (float*)d_out;
    float* out_boxes = out;                      // NB*4 floats
    float* out_cls   = out + NB * 4;             // NB floats
    float* out_valid = out + NB * 5;             // NB floats

    const int grid = (int)((NB + TILE - 1) / TILE);  // 4366 blocks
    ssd_decode_kernel<<<grid, TILE, 0, stream>>>(off, conf, dbox,
                                                 out_boxes, out_cls, out_valid);
}